// GNNEncoder_90383291777088
// MI455X (gfx1250) — compile-verified
//
#include <hip/hip_runtime.h>
#include <stdint.h>

#define NN      100000
#define EE      1600000
#define EPL     (EE + NN)        // edges + self loops
#define IN_DIM  384
#define HID     256
#define NH      4
#define CHD     64
#define SLOPE   0.2f

typedef __attribute__((ext_vector_type(2))) float v2f;
typedef __attribute__((ext_vector_type(8))) float v8f;

// ---- order-preserving uint encoding of float (for atomicMax on f32) ----
__device__ __forceinline__ unsigned ord_f32(float f) {
  unsigned b = __float_as_uint(f);
  return (b & 0x80000000u) ? ~b : (b | 0x80000000u);
}
__device__ __forceinline__ float unord_f32(unsigned u) {
  unsigned b = (u & 0x80000000u) ? (u ^ 0x80000000u) : ~u;
  return __uint_as_float(b);
}

// =====================================================================
// GEMM: C[M,256] = act(A[M,K] @ B[K,256] + bias), fp32 WMMA 16x16x4.
// One wave computes a 16x256 tile (16 v8f accumulators) -> A read once.
// A frag (ISA 32-bit A 16x4): lanes0-15 rowM=lid K={k,k+1}; lanes16-31 K={k+2,k+3}
// B frag mirrored; C/D: vgpr r -> row = 8*half + r, col = lane&15 within subtile.
// =====================================================================
__global__ void __launch_bounds__(128)
gemm_wmma_n256(const float* __restrict__ A, const float* __restrict__ B,
               const float* __restrict__ bias, float* __restrict__ C,
               int M, int K, int act) {
  const int lane = threadIdx.x & 31;
  const int wave = blockIdx.x * (blockDim.x >> 5) + (threadIdx.x >> 5);
  if (wave * 16 >= M) return;
  const int lid  = lane & 15;
  const int half = lane >> 4;
  const float* __restrict__ arow = A + (size_t)(wave * 16 + lid) * K;

  v8f acc[16];
#pragma unroll
  for (int j = 0; j < 16; ++j) acc[j] = (v8f)(0.0f);

  for (int k = 0; k < K; k += 4) {
    const int ka = k + (half << 1);
    v2f a;
    a.x = arow[ka];
    a.y = arow[ka + 1];
    const float* __restrict__ b0 = B + (size_t)ka * 256 + lid;
#pragma unroll
    for (int j = 0; j < 16; ++j) {
      v2f b;
      b.x = b0[j * 16];
      b.y = b0[j * 16 + 256];
      acc[j] = __builtin_amdgcn_wmma_f32_16x16x4_f32(
          false, a, false, b, (short)0, acc[j], false, false);
    }
  }

  const int rbase = wave * 16 + half * 8;
#pragma unroll
  for (int j = 0; j < 16; ++j) {
    const int col = j * 16 + lid;
    const float bv = bias ? bias[col] : 0.0f;
#pragma unroll
    for (int r = 0; r < 8; ++r) {
      float v = acc[j][r] + bv;
      if (act) v = fmaxf(v, 0.0f);
      C[(size_t)(rbase + r) * 256 + col] = v;
    }
  }
}

// ---- edge list: int64 [2,E] -> int32 src/dst with self loops appended ----
__global__ void build_edges(const long long* __restrict__ ei,
                            int* __restrict__ src, int* __restrict__ dst) {
  int e = blockIdx.x * blockDim.x + threadIdx.x;
  if (e >= EPL) return;
  if (e < EE) { src[e] = (int)ei[e]; dst[e] = (int)ei[EE + e]; }
  else        { src[e] = e - EE;     dst[e] = e - EE; }
}

// ---- per-node attention scores: a_src[n,h] = <hlin[n,h,:], asrc[h,:]> ----
__global__ void node_scores(const float* __restrict__ hlin,
                            const float* __restrict__ asrc,
                            const float* __restrict__ adst,
                            float* __restrict__ a_src, float* __restrict__ a_dst) {
  int idx = blockIdx.x * blockDim.x + threadIdx.x;
  if (idx >= NN * NH) return;
  int n = idx >> 2, h = idx & 3;
  const float4* hp = (const float4*)(hlin + (size_t)n * HID + h * CHD);
  const float4* ws = (const float4*)(asrc + h * CHD);
  const float4* wd = (const float4*)(adst + h * CHD);
  float ss = 0.f, sd = 0.f;
#pragma unroll
  for (int i = 0; i < 16; ++i) {
    float4 v = hp[i], s = ws[i], d = wd[i];
    ss += v.x * s.x + v.y * s.y + v.z * s.z + v.w * s.w;
    sd += v.x * d.x + v.y * d.y + v.z * d.z + v.w * d.w;
  }
  a_src[idx] = ss;
  a_dst[idx] = sd;
}

// ---- pass 1: segment max of leaky_relu(a_src[s]+a_dst[d]) over dst ----
__global__ void edge_max(const int* __restrict__ src, const int* __restrict__ dst,
                         const float* __restrict__ a_src, const float* __restrict__ a_dst,
                         unsigned* __restrict__ amax_u) {
  int e = blockIdx.x * blockDim.x + threadIdx.x;
  if (e >= EPL) return;
  int s = src[e], d = dst[e];
  float4 as = *(const float4*)(a_src + (size_t)s * 4);
  float4 ad = *(const float4*)(a_dst + (size_t)d * 4);
  float al[4] = {as.x + ad.x, as.y + ad.y, as.z + ad.z, as.w + ad.w};
#pragma unroll
  for (int h = 0; h < 4; ++h) {
    float v = al[h] > 0.f ? al[h] : SLOPE * al[h];
    atomicMax(amax_u + (size_t)d * 4 + h, ord_f32(v));
  }
}

// ---- pass 2: ealpha = exp(alpha - max[dst]); denom[dst] += ealpha ----
__global__ void edge_expsum(const int* __restrict__ src, const int* __restrict__ dst,
                            const float* __restrict__ a_src, const float* __restrict__ a_dst,
                            const unsigned* __restrict__ amax_u,
                            float* __restrict__ ealpha, float* __restrict__ denom) {
  int e = blockIdx.x * blockDim.x + threadIdx.x;
  if (e >= EPL) return;
  int s = src[e], d = dst[e];
  float4 as = *(const float4*)(a_src + (size_t)s * 4);
  float4 ad = *(const float4*)(a_dst + (size_t)d * 4);
  float al[4] = {as.x + ad.x, as.y + ad.y, as.z + ad.z, as.w + ad.w};
  float4 out;
  float* op = (float*)&out;
#pragma unroll
  for (int h = 0; h < 4; ++h) {
    float v = al[h] > 0.f ? al[h] : SLOPE * al[h];
    float mx = unord_f32(amax_u[(size_t)d * 4 + h]);
    float ev = expf(v - mx);
    op[h] = ev;
    atomicAdd(denom + (size_t)d * 4 + h, ev);
  }
  *(float4*)(ealpha + (size_t)e * 4) = out;
}

// ---- pass 3: agg[dst] += hlin[src] * (ealpha/denom[dst]); wave per edge ----
__global__ void __launch_bounds__(256)
edge_aggregate(const int* __restrict__ src, const int* __restrict__ dst,
               const float* __restrict__ hlin, const float* __restrict__ ealpha,
               const float* __restrict__ denom, float* __restrict__ agg) {
  int e = blockIdx.x * 8 + (threadIdx.x >> 5);
  if (e >= EPL) return;
  int lane = threadIdx.x & 31;
  int s = src[e], d = dst[e];
  int h = lane >> 3;                       // 8 channels per lane, all in one head
  float w = ealpha[(size_t)e * 4 + h] / denom[(size_t)d * 4 + h];
  const float4* hp = (const float4*)(hlin + (size_t)s * HID + lane * 8);
  float4 v0 = hp[0], v1 = hp[1];
  float* ap = agg + (size_t)d * HID + lane * 8;
  atomicAdd(ap + 0, v0.x * w);
  atomicAdd(ap + 1, v0.y * w);
  atomicAdd(ap + 2, v0.z * w);
  atomicAdd(ap + 3, v0.w * w);
  atomicAdd(ap + 4, v1.x * w);
  atomicAdd(ap + 5, v1.y * w);
  atomicAdd(ap + 6, v1.z * w);
  atomicAdd(ap + 7, v1.w * w);
}

// ---- h += elu(agg + bias) ----
__global__ void finalize_layer(const float* __restrict__ agg,
                               const float* __restrict__ bias,
                               float* __restrict__ h) {
  int i = blockIdx.x * blockDim.x + threadIdx.x;
  if (i >= NN * HID) return;
  float v = agg[i] + bias[i & 255];
  v = v > 0.f ? v : (expf(v) - 1.0f);
  h[i] += v;
}

// =====================================================================
extern "C" void kernel_launch(void* const* d_in, const int* in_sizes, int n_in,
                              void* d_out, int out_size, void* d_ws, size_t ws_size,
                              hipStream_t stream) {
  (void)in_sizes; (void)n_in; (void)out_size; (void)ws_size;

  const float*     x    = (const float*)d_in[0];
  const long long* ei   = (const long long*)d_in[1];
  const float*     w_in = (const float*)d_in[2];
  const float*     b_in = (const float*)d_in[3];
  const float*     lin[2]  = {(const float*)d_in[4],  (const float*)d_in[8]};
  const float*     asr[2]  = {(const float*)d_in[5],  (const float*)d_in[9]};
  const float*     ads[2]  = {(const float*)d_in[6],  (const float*)d_in[10]};
  const float*     bia[2]  = {(const float*)d_in[7],  (const float*)d_in[11]};
  const float*     w_out   = (const float*)d_in[12];
  const float*     b_out   = (const float*)d_in[13];
  float* out = (float*)d_out;

  // workspace carve-up (floats)
  float* ws = (float*)d_ws;
  size_t o = 0;
  float*    h      = ws + o; o += (size_t)NN * HID;   // 25.6M
  float*    hlin   = ws + o; o += (size_t)NN * HID;
  float*    agg    = ws + o; o += (size_t)NN * HID;
  float*    a_src  = ws + o; o += (size_t)NN * NH;
  float*    a_dst  = ws + o; o += (size_t)NN * NH;
  unsigned* amax_u = (unsigned*)(ws + o); o += (size_t)NN * NH;
  float*    denom  = ws + o; o += (size_t)NN * NH;
  float*    ealpha = ws + o; o += (size_t)EPL * NH;
  int*      srcI   = (int*)(ws + o); o += (size_t)EPL;
  int*      dstI   = (int*)(ws + o); o += (size_t)EPL;

  const int TPB = 256;
  const int gemm_blocks = ((NN / 16) + 3) / 4;      // 4 waves/block, 16 rows/wave
  const int edge_blocks = (EPL + TPB - 1) / TPB;
  const int node_blocks = (NN * NH + TPB - 1) / TPB;
  const int feat_blocks = (NN * HID + TPB - 1) / TPB;
  const int aggr_blocks = (EPL + 7) / 8;            // wave per edge, 8 waves/block

  // edge compaction (+ self loops)
  build_edges<<<edge_blocks, TPB, 0, stream>>>(ei, srcI, dstI);

  // input projection + relu
  gemm_wmma_n256<<<gemm_blocks, 128, 0, stream>>>(x, w_in, b_in, h, NN, IN_DIM, 1);

  for (int L = 0; L < 2; ++L) {
    // hlin = h @ lin
    gemm_wmma_n256<<<gemm_blocks, 128, 0, stream>>>(h, lin[L], nullptr, hlin, NN, HID, 0);
    // per-node attention scores
    node_scores<<<node_blocks, TPB, 0, stream>>>(hlin, asr[L], ads[L], a_src, a_dst);
    // reset reduction buffers (capturable memset nodes)
    hipMemsetAsync(amax_u, 0, (size_t)NN * NH * sizeof(unsigned), stream);
    hipMemsetAsync(denom,  0, (size_t)NN * NH * sizeof(float),    stream);
    hipMemsetAsync(agg,    0, (size_t)NN * HID * sizeof(float),   stream);
    // segment softmax + weighted scatter
    edge_max<<<edge_blocks, TPB, 0, stream>>>(srcI, dstI, a_src, a_dst, amax_u);
    edge_expsum<<<edge_blocks, TPB, 0, stream>>>(srcI, dstI, a_src, a_dst, amax_u, ealpha, denom);
    edge_aggregate<<<aggr_blocks, TPB, 0, stream>>>(srcI, dstI, hlin, ealpha, denom, agg);
    // h += elu(agg + bias)
    finalize_layer<<<feat_blocks, TPB, 0, stream>>>(agg, bia[L], h);
  }

  // output projection
  gemm_wmma_n256<<<gemm_blocks, 128, 0, stream>>>(h, w_out, b_out, out, NN, HID, 0);
}